// Model_26903675142176
// MI455X (gfx1250) — compile-verified
//
#include <hip/hip_runtime.h>
#include <hip/hip_bf16.h>

#define BS   4096
#define HID  1024
#define HALF 512

typedef float v2f __attribute__((ext_vector_type(2)));
typedef float v8f __attribute__((ext_vector_type(8)));
typedef int   v4i __attribute__((vector_size(16)));

#define AS1 __attribute__((address_space(1)))
#define AS3 __attribute__((address_space(3)))

#if __has_builtin(__builtin_amdgcn_global_load_async_to_lds_b128) && \
    __has_builtin(__builtin_amdgcn_s_wait_asynccnt)
#define ASYNC_OK 1
#else
#define ASYNC_OK 0
#endif

__device__ __forceinline__ v8f wmma4(v2f a, v2f b, v8f c) {
    // D = A(16x4 f32) * B(4x16 f32) + C(16x16 f32)
    return __builtin_amdgcn_wmma_f32_16x16x4_f32(false, a, false, b, (short)0, c, false, false);
}

#if ASYNC_OK
__device__ __forceinline__ void asyncCopy16(const float* gp, float* lp) {
    // global -> LDS, 16 bytes, tracked by ASYNCcnt (no VGPR round-trip)
    __builtin_amdgcn_global_load_async_to_lds_b128(
        (AS1 v4i*)(uintptr_t)gp,
        (AS3 v4i*)(unsigned)(uintptr_t)lp, 0, 0);
}
template <int N> __device__ __forceinline__ void waitAsync() {
    __builtin_amdgcn_s_wait_asynccnt((unsigned short)N);
}
#endif

__device__ __forceinline__ float waveSum(float v) {
    #pragma unroll
    for (int off = 16; off > 0; off >>= 1) v += __shfl_down(v, off, 32);
    return v;
}

__global__ void zero3_kernel(float* out) {
    if (threadIdx.x < 3) out[threadIdx.x] = 0.0f;
}

// prob_1[i] = dot(o0[i], o1[i]); prob_2[i] = dot(o0[i], o2[i])
__global__ void dot12_kernel(const float* __restrict__ outs,
                             float* __restrict__ p1, float* __restrict__ p2) {
    int wave = threadIdx.x >> 5, lane = threadIdx.x & 31;
    int row = blockIdx.x * 8 + wave;
    const float* a = outs + (size_t)row * HID;
    const float* b = outs + (size_t)(BS + row) * HID;
    const float* c = outs + (size_t)(2 * BS + row) * HID;
    float s1 = 0.f, s2 = 0.f;
    for (int k = lane; k < HID; k += 32) {
        float av = a[k];
        s1 += av * b[k];
        s2 += av * c[k];
    }
    s1 = waveSum(s1); s2 = waveSum(s2);
    if (lane == 0) { p1[row] = s1; p2[row] = s2; }
}

// ---- WMMA tile compute helpers ----------------------------------------------

// 64x64 GEMM tile: wave computes 32x32 (2x2 frags). As[row][k], Bs[k][n].
__device__ __forceinline__ void mmaTile64(const float (&As)[64][36], const float (&Bs)[32][64],
                                          int ra, int cb, int halfK, v8f acc[2][2]) {
    v2f aF[2][8];
    #pragma unroll
    for (int kk = 0; kk < 8; ++kk) {
        int kb = kk * 4 + halfK * 2;
        aF[0][kk].x = As[ra][kb];      aF[0][kk].y = As[ra][kb + 1];
        aF[1][kk].x = As[ra + 16][kb]; aF[1][kk].y = As[ra + 16][kb + 1];
    }
    #pragma unroll
    for (int kk = 0; kk < 8; ++kk) {
        int kb = kk * 4 + halfK * 2;
        v2f b0, b1;
        b0.x = Bs[kb][cb];      b0.y = Bs[kb + 1][cb];
        b1.x = Bs[kb][cb + 16]; b1.y = Bs[kb + 1][cb + 16];
        acc[0][0] = wmma4(aF[0][kk], b0, acc[0][0]);
        acc[0][1] = wmma4(aF[0][kk], b1, acc[0][1]);
        acc[1][0] = wmma4(aF[1][kk], b0, acc[1][0]);
        acc[1][1] = wmma4(aF[1][kk], b1, acc[1][1]);
    }
}

// 16x64 similarity tile: wave owns 16 rows x 4 col-frags. Both operands [row][k].
__device__ __forceinline__ void mmaTileCtr(const float (&As)[64][36], const float (&Bs)[64][36],
                                           int rr, int l16, int halfK, v8f acc[4]) {
    v2f aF[8];
    #pragma unroll
    for (int kk = 0; kk < 8; ++kk) {
        int kb = kk * 4 + halfK * 2;
        aF[kk].x = As[rr][kb]; aF[kk].y = As[rr][kb + 1];
    }
    #pragma unroll
    for (int kk = 0; kk < 8; ++kk) {
        int kb = kk * 4 + halfK * 2;
        #pragma unroll
        for (int f = 0; f < 4; ++f) {
            int cc = f * 16 + l16;
            v2f b; b.x = Bs[cc][kb]; b.y = Bs[cc][kb + 1];
            acc[f] = wmma4(aF[kk], b, acc[f]);
        }
    }
}

// ---- contrastive loss: fused flash-softmax over [p1, p2, masked(o0 @ temp^T)]
__global__ void ctr_loss_kernel(const float* __restrict__ outs,
                                const int* __restrict__ labels,
                                const float* __restrict__ p1arr,
                                const float* __restrict__ p2arr,
                                float* __restrict__ out3) {
    __shared__ float As[2][64][36];
    __shared__ float Bs[2][64][36];
    __shared__ float Sld[64][65];
    __shared__ int   rowLab[64];

    int tid = threadIdx.x;             // 128 threads
    int wave = tid >> 5, lane = tid & 31;
    int halfK = lane >> 4, l16 = lane & 15;
    int r0 = blockIdx.x * 64;
    int rr = wave * 16 + l16;

    if (tid < 64) rowLab[tid] = labels[r0 + tid];

    float m = -3.0e38f, s = 0.0f;
    if (tid < 64) {
        float p1 = p1arr[r0 + tid], p2 = p2arr[r0 + tid];
        m = fmaxf(p1, p2);
        s = expf(p1 - m) + expf(p2 - m);
    }
    __syncthreads();

    for (int n0 = 0; n0 < 2 * BS; n0 += 64) {
        v8f acc[4] = {};
#if ASYNC_OK
        #pragma unroll
        for (int i = 0; i < 4; ++i) {
            int sx = tid + i * 128; int r = sx >> 3, c = (sx & 7) * 4;
            asyncCopy16(&outs[(size_t)(r0 + r) * HID + c], &As[0][r][c]);
            asyncCopy16(&outs[(size_t)(n0 + r) * HID + c], &Bs[0][r][c]);
        }
        for (int kt = 0; kt < HID; kt += 32) {
            int p = (kt >> 5) & 1;
            if (kt + 32 < HID) {
                #pragma unroll
                for (int i = 0; i < 4; ++i) {
                    int sx = tid + i * 128; int r = sx >> 3, c = (sx & 7) * 4;
                    asyncCopy16(&outs[(size_t)(r0 + r) * HID + kt + 32 + c], &As[1 - p][r][c]);
                    asyncCopy16(&outs[(size_t)(n0 + r) * HID + kt + 32 + c], &Bs[1 - p][r][c]);
                }
                waitAsync<8>();
            } else {
                waitAsync<0>();
            }
            __syncthreads();
            mmaTileCtr(As[p], Bs[p], rr, l16, halfK, acc);
            __syncthreads();
        }
#else
        float4 rga[4], rgb[4];
        #pragma unroll
        for (int i = 0; i < 4; ++i) {
            int sx = tid + i * 128; int r = sx >> 3, c = (sx & 7) * 4;
            rga[i] = *(const float4*)&outs[(size_t)(r0 + r) * HID + c];
            rgb[i] = *(const float4*)&outs[(size_t)(n0 + r) * HID + c];
        }
        #pragma unroll
        for (int i = 0; i < 4; ++i) {
            int sx = tid + i * 128; int r = sx >> 3, c = (sx & 7) * 4;
            *(float4*)&As[0][r][c] = rga[i];
            *(float4*)&Bs[0][r][c] = rgb[i];
        }
        __syncthreads();
        for (int kt = 0; kt < HID; kt += 32) {
            int p = (kt >> 5) & 1;
            bool nxt = (kt + 32 < HID);
            if (nxt) {
                #pragma unroll
                for (int i = 0; i < 4; ++i) {
                    int sx = tid + i * 128; int r = sx >> 3, c = (sx & 7) * 4;
                    rga[i] = *(const float4*)&outs[(size_t)(r0 + r) * HID + kt + 32 + c];
                    rgb[i] = *(const float4*)&outs[(size_t)(n0 + r) * HID + kt + 32 + c];
                }
            }
            mmaTileCtr(As[p], Bs[p], rr, l16, halfK, acc);
            __syncthreads();
            if (nxt) {
                #pragma unroll
                for (int i = 0; i < 4; ++i) {
                    int sx = tid + i * 128; int r = sx >> 3, c = (sx & 7) * 4;
                    *(float4*)&As[1 - p][r][c] = rga[i];
                    *(float4*)&Bs[1 - p][r][c] = rgb[i];
                }
                __syncthreads();
            }
        }
#endif
        // Mask (labels[i]==temp_labels[j] -> -1e9) and spill S tile to LDS.
        #pragma unroll
        for (int f = 0; f < 4; ++f) {
            #pragma unroll
            for (int v = 0; v < 8; ++v) {
                int row  = wave * 16 + v + halfK * 8;
                int colL = f * 16 + l16;
                int j = n0 + colL;
                float val = acc[f][v];
                if (rowLab[row] == labels[j & (BS - 1)]) val = -1.0e9f;
                Sld[row][colL] = val;
            }
        }
        __syncthreads();
        if (tid < 64) {
            float mt = -3.0e38f;
            #pragma unroll 8
            for (int j = 0; j < 64; ++j) mt = fmaxf(mt, Sld[tid][j]);
            float nm = fmaxf(m, mt);
            float a2 = s * expf(m - nm);
            #pragma unroll 8
            for (int j = 0; j < 64; ++j) a2 += expf(Sld[tid][j] - nm);
            m = nm; s = a2;
        }
        __syncthreads();
    }
    if (tid < 64) {
        float p1 = p1arr[r0 + tid];
        float p0 = expf(p1 - m) / s;
        atomicAdd(&out3[0], -logf(p0 + 1e-10f) * (1.0f / (float)BS));
    }
}

// ---- C[M,N] = act(A[M,K] @ W[K,N] + bias), 64x64 block, 4 waves x (32x32) ----
template <int RELU>
__global__ void gemm_bias_kernel(const float* __restrict__ A,
                                 const float* __restrict__ W,
                                 const float* __restrict__ bias,
                                 float* __restrict__ C, int M, int N, int K) {
    __shared__ float As[2][64][36];
    __shared__ float Bs[2][32][64];
    int tid = threadIdx.x;             // 128
    int wave = tid >> 5, lane = tid & 31;
    int wm = wave & 1, wn = wave >> 1;
    int halfK = lane >> 4, l16 = lane & 15;
    int n0 = blockIdx.x * 64, m0 = blockIdx.y * 64;
    int ra = wm * 32 + l16;
    int cb = wn * 32 + l16;

    v8f acc[2][2] = {};

#if ASYNC_OK
    #pragma unroll
    for (int i = 0; i < 4; ++i) {
        int sx = tid + i * 128;
        int r = sx >> 3, c = (sx & 7) * 4;
        asyncCopy16(&A[(size_t)(m0 + r) * K + c], &As[0][r][c]);
        int kr = sx >> 4, c2 = (sx & 15) * 4;
        asyncCopy16(&W[(size_t)kr * N + n0 + c2], &Bs[0][kr][c2]);
    }
    for (int kt = 0; kt < K; kt += 32) {
        int p = (kt >> 5) & 1;
        if (kt + 32 < K) {
            #pragma unroll
            for (int i = 0; i < 4; ++i) {
                int sx = tid + i * 128;
                int r = sx >> 3, c = (sx & 7) * 4;
                asyncCopy16(&A[(size_t)(m0 + r) * K + kt + 32 + c], &As[1 - p][r][c]);
                int kr = sx >> 4, c2 = (sx & 15) * 4;
                asyncCopy16(&W[(size_t)(kt + 32 + kr) * N + n0 + c2], &Bs[1 - p][kr][c2]);
            }
            waitAsync<8>();
        } else {
            waitAsync<0>();
        }
        __syncthreads();
        mmaTile64(As[p], Bs[p], ra, cb, halfK, acc);
        __syncthreads();
    }
#else
    float4 rga[4], rgb[4];
    #pragma unroll
    for (int i = 0; i < 4; ++i) {
        int sx = tid + i * 128;
        int r = sx >> 3, c = (sx & 7) * 4;
        rga[i] = *(const float4*)&A[(size_t)(m0 + r) * K + c];
        int kr = sx >> 4, c2 = (sx & 15) * 4;
        rgb[i] = *(const float4*)&W[(size_t)kr * N + n0 + c2];
    }
    #pragma unroll
    for (int i = 0; i < 4; ++i) {
        int sx = tid + i * 128;
        int r = sx >> 3, c = (sx & 7) * 4;
        *(float4*)&As[0][r][c] = rga[i];
        int kr = sx >> 4, c2 = (sx & 15) * 4;
        *(float4*)&Bs[0][kr][c2] = rgb[i];
    }
    __syncthreads();
    for (int kt = 0; kt < K; kt += 32) {
        int p = (kt >> 5) & 1;
        bool nxt = (kt + 32 < K);
        if (nxt) {
            #pragma unroll
            for (int i = 0; i < 4; ++i) {
                int sx = tid + i * 128;
                int r = sx >> 3, c = (sx & 7) * 4;
                rga[i] = *(const float4*)&A[(size_t)(m0 + r) * K + kt + 32 + c];
                int kr = sx >> 4, c2 = (sx & 15) * 4;
                rgb[i] = *(const float4*)&W[(size_t)(kt + 32 + kr) * N + n0 + c2];
            }
        }
        mmaTile64(As[p], Bs[p], ra, cb, halfK, acc);
        __syncthreads();
        if (nxt) {
            #pragma unroll
            for (int i = 0; i < 4; ++i) {
                int sx = tid + i * 128;
                int r = sx >> 3, c = (sx & 7) * 4;
                *(float4*)&As[1 - p][r][c] = rga[i];
                int kr = sx >> 4, c2 = (sx & 15) * 4;
                *(float4*)&Bs[1 - p][kr][c2] = rgb[i];
            }
            __syncthreads();
        }
    }
#endif

    #pragma unroll
    for (int i = 0; i < 2; ++i) {
        #pragma unroll
        for (int j = 0; j < 2; ++j) {
            int col = n0 + wn * 32 + j * 16 + l16;
            float bv = bias[col];
            #pragma unroll
            for (int v = 0; v < 8; ++v) {
                int row = m0 + wm * 32 + i * 16 + v + halfK * 8;
                float x = acc[i][j][v] + bv;
                if (RELU) x = fmaxf(x, 0.0f);
                C[(size_t)row * N + col] = x;
            }
        }
    }
}

// fv_raw = is_pj ? pj : jp
__global__ void select_fv_kernel(const float* __restrict__ pj,
                                 const float* __restrict__ jp,
                                 const int* __restrict__ dom,
                                 float* __restrict__ fv) {
    size_t idx = (size_t)blockIdx.x * 256 + threadIdx.x;
    int row = (int)(idx >> 10);               // /HID
    int dl = dom[row & (BS - 1)];
    fv[idx] = (dl == 1) ? pj[idx] : jp[idx];
}

// cycle_loss = mean |outputss - cycle_norm| ; wave-per-row
__global__ void cycle_loss_kernel(const float* __restrict__ outs,
                                  const float* __restrict__ cycJ,
                                  const float* __restrict__ cycP,
                                  const int* __restrict__ dom,
                                  float* __restrict__ out3) {
    int row = blockIdx.x * 8 + (threadIdx.x >> 5);
    int lane = threadIdx.x & 31;
    int dl = dom[row & (BS - 1)];
    const float* cyc = ((dl == 1) ? cycJ : cycP) + (size_t)row * HID;
    const float* o = outs + (size_t)row * HID;
    float sq = 0.f;
    for (int k = lane; k < HID; k += 32) { float v = cyc[k]; sq += v * v; }
    sq = waveSum(sq);
    sq = __shfl(sq, 0, 32);
    float inv = 1.0f / (sqrtf(sq) + 1e-13f);
    float s = 0.f;
    for (int k = lane; k < HID; k += 32) s += fabsf(o[k] - cyc[k] * inv);
    s = waveSum(s);
    if (lane == 0)
        atomicAdd(&out3[2], s * (1.0f / ((float)(3 * BS) * (float)HID)));
}

// One segment of the domain cross-entropy; normalization folded into the dot.
__global__ void domain_kernel(const float* __restrict__ src,
                              const float* __restrict__ srcAlt,
                              int useSelect, int useNorm, int flip,
                              const int* __restrict__ dom,
                              const float* __restrict__ wdom,
                              const float* __restrict__ bdom,
                              float* __restrict__ out3) {
    int row = blockIdx.x * 8 + (threadIdx.x >> 5);
    int lane = threadIdx.x & 31;
    int dl = dom[row & (BS - 1)];
    const float* v = src;
    if (useSelect && dl != 1) v = srcAlt;
    v += (size_t)row * HID;
    float d0 = 0.f, d1 = 0.f, sq = 0.f;
    for (int k = lane; k < HID; k += 32) {
        float x = v[k];
        d0 += x * wdom[2 * k];
        d1 += x * wdom[2 * k + 1];
        sq += x * x;
    }
    d0 = waveSum(d0); d1 = waveSum(d1); sq = waveSum(sq);
    if (lane == 0) {
        float sc = useNorm ? 1.0f / (sqrtf(sq) + 1e-13f) : 1.0f;
        float z0 = d0 * sc + bdom[0];
        float z1 = d1 * sc + bdom[1];
        float mx = fmaxf(z0, z1);
        float lse = mx + logf(expf(z0 - mx) + expf(z1 - mx));
        int t = flip ? (1 - dl) : dl;
        float lp = (t ? z1 : z0) - lse;
        atomicAdd(&out3[1], -lp * (1.0f / (9.0f * (float)BS)));
    }
}

extern "C" void kernel_launch(void* const* d_in, const int* in_sizes, int n_in,
                              void* d_out, int out_size, void* d_ws, size_t ws_size,
                              hipStream_t stream) {
    (void)in_sizes; (void)n_in; (void)out_size; (void)ws_size;
    const float* outs  = (const float*)d_in[0];
    const int*   labels = (const int*)d_in[1];
    // d_in[2] negative_labels, d_in[4] alpha: unused in forward
    const int*   dom   = (const int*)d_in[3];
    const float* Wpj1 = (const float*)d_in[5];
    const float* bpj1 = (const float*)d_in[6];
    const float* Wpj2 = (const float*)d_in[7];
    const float* bpj2 = (const float*)d_in[8];
    const float* Wjp1 = (const float*)d_in[9];
    const float* bjp1 = (const float*)d_in[10];
    const float* Wjp2 = (const float*)d_in[11];
    const float* bjp2 = (const float*)d_in[12];
    const float* Wdom = (const float*)d_in[13];
    const float* bdom = (const float*)d_in[14];
    float* out = (float*)d_out;

    // Workspace: bufA, bufB, fv (each 3*BS*HID), hid (3*BS*HALF), p1, p2. ~176 MB.
    float* w = (float*)d_ws;
    const size_t R = (size_t)3 * BS * HID;
    float* bufA = w;
    float* bufB = w + R;
    float* fv   = w + 2 * R;
    float* hid  = w + 3 * R;
    float* p1   = hid + (size_t)3 * BS * HALF;
    float* p2   = p1 + BS;

    zero3_kernel<<<1, 32, 0, stream>>>(out);
    // o0 output copy (4th return value)
    (void)hipMemcpyAsync(out + 3, outs, (size_t)BS * HID * sizeof(float),
                         hipMemcpyDeviceToDevice, stream);

    // Contrastive loss
    dot12_kernel<<<BS / 8, 256, 0, stream>>>(outs, p1, p2);
    ctr_loss_kernel<<<BS / 64, 128, 0, stream>>>(outs, labels, p1, p2, out);

    dim3 g1(HALF / 64, (3 * BS) / 64);   // N=512
    dim3 g2(HID / 64, (3 * BS) / 64);    // N=1024

    // pj = mlp_pj(outputss), jp = mlp_jp(outputss)
    gemm_bias_kernel<1><<<g1, 128, 0, stream>>>(outs, Wpj1, bpj1, hid, 3 * BS, HALF, HID);
    gemm_bias_kernel<0><<<g2, 128, 0, stream>>>(hid, Wpj2, bpj2, bufA, 3 * BS, HID, HALF);
    gemm_bias_kernel<1><<<g1, 128, 0, stream>>>(outs, Wjp1, bjp1, hid, 3 * BS, HALF, HID);
    gemm_bias_kernel<0><<<g2, 128, 0, stream>>>(hid, Wjp2, bjp2, bufB, 3 * BS, HID, HALF);

    // fv_raw = where(is_pj, pj, jp)
    select_fv_kernel<<<(3 * BS * HID) / 256, 256, 0, stream>>>(bufA, bufB, dom, fv);

    // cycJ = mlp_jp(fv) -> bufA (pj dead), cycP = mlp_pj(fv) -> bufB (jp dead)
    gemm_bias_kernel<1><<<g1, 128, 0, stream>>>(fv, Wjp1, bjp1, hid, 3 * BS, HALF, HID);
    gemm_bias_kernel<0><<<g2, 128, 0, stream>>>(hid, Wjp2, bjp2, bufA, 3 * BS, HID, HALF);
    gemm_bias_kernel<1><<<g1, 128, 0, stream>>>(fv, Wpj1, bpj1, hid, 3 * BS, HALF, HID);
    gemm_bias_kernel<0><<<g2, 128, 0, stream>>>(hid, Wpj2, bpj2, bufB, 3 * BS, HID, HALF);

    // cycle loss (normalizes selected cycle rows on the fly)
    cycle_loss_kernel<<<(3 * BS) / 8, 256, 0, stream>>>(outs, bufA, bufB, dom, out);

    // domain loss: seg0 raw outputss, seg1 fv normalized (target flipped),
    // seg2 selected cycle normalized
    domain_kernel<<<(3 * BS) / 8, 256, 0, stream>>>(outs, outs, 0, 0, 0, dom, Wdom, bdom, out);
    domain_kernel<<<(3 * BS) / 8, 256, 0, stream>>>(fv, fv, 0, 1, 1, dom, Wdom, bdom, out);
    domain_kernel<<<(3 * BS) / 8, 256, 0, stream>>>(bufA, bufB, 1, 1, 0, dom, Wdom, bdom, out);
}